// SAGE_38345468019266
// MI455X (gfx1250) — compile-verified
//
#include <hip/hip_runtime.h>

#define N_NODES 100000
#define N_EDGES 1600000
#define EMB 128
#define AGGD 256            // 2*EMB aggregate width
#define ZDIM 384            // 3*EMB logical GEMM K
#define ZPAD 388            // LDS row stride (384%64==0 would 16-way bank conflict)
#define MTILE 32            // rows per block in GEMM (100000 = 32*3125)

typedef __attribute__((ext_vector_type(2))) float v2f;
typedef __attribute__((ext_vector_type(8))) float v8f;

// ---------------- small utility kernels ----------------

__global__ void zero_f(float* __restrict__ p, int n) {
    int i = blockIdx.x * blockDim.x + threadIdx.x;
    if (i < n) p[i] = 0.0f;
}

__global__ void copy_f(float* __restrict__ dst, const float* __restrict__ src, int n) {
    int i = blockIdx.x * blockDim.x + threadIdx.x;
    if (i < n) dst[i] = src[i];
}

// in-degree counts (float, for mean divide)
__global__ void count_deg(float* __restrict__ counts, const int* __restrict__ dst, int e) {
    int i = blockIdx.x * blockDim.x + threadIdx.x;
    if (i < e) atomicAdd(&counts[dst[i]], 1.0f);
}

// counts -> 1/max(counts,1), in place
__global__ void inv_denom(float* __restrict__ counts, int n) {
    int i = blockIdx.x * blockDim.x + threadIdx.x;
    if (i < n) {
        float c = counts[i];
        counts[i] = 1.0f / (c > 1.0f ? c : 1.0f);
    }
}

// 4 edges per 256-thread block; 64 lanes per edge, float4 per lane.
// agg[dst][0:128) += h_in[src]; agg[dst][128:256) += h_r[e]
__global__ __launch_bounds__(256) void scatter_edges(float* __restrict__ agg,
                                                     const float* __restrict__ h_in,
                                                     const float* __restrict__ h_r,
                                                     const int* __restrict__ src,
                                                     const int* __restrict__ dst) {
    const int e  = blockIdx.x * 4 + (threadIdx.x >> 6);
    const int t4 = threadIdx.x & 63;                 // owns cols [4*t4, 4*t4+4)
    const int s  = src[e];
    const int d  = dst[e];
    const float4* p = (t4 < 32)
        ? (const float4*)(h_in + (size_t)s * EMB) + t4
        : (const float4*)(h_r  + (size_t)e * EMB) + (t4 - 32);
    float4 v = *p;                                   // global_load_b128, coalesced
    float* q = agg + (size_t)d * AGGD + 4 * t4;
    atomicAdd(q + 0, v.x);                           // global_atomic_add_f32; agg ~102MB,
    atomicAdd(q + 1, v.y);                           // resolves mostly in the 192MB L2
    atomicAdd(q + 2, v.z);
    atomicAdd(q + 3, v.w);
}

// ------- WMMA f32 GEMM: out = relu([h_in | agg*invd] @ W[384,128] + b) -------
// grid.x = N/32; 256 threads = 8 waves; wave w owns cols [16w,16w+16), M-tile = 32.
__global__ __launch_bounds__(256) void gemm_bias_relu(const float* __restrict__ h_in,
                                                      const float* __restrict__ agg,
                                                      const float* __restrict__ invd,
                                                      const float* __restrict__ W,
                                                      const float* __restrict__ bias,
                                                      float* __restrict__ out) {
    __shared__ float zt[MTILE * ZPAD];

    const int rowBase = blockIdx.x * MTILE;

    // stage 32x384 z tile: self cols from h_in, agg cols pre-scaled by 1/denom
    for (int i = threadIdx.x; i < MTILE * ZDIM; i += 256) {
        int r = i / ZDIM;
        int c = i - r * ZDIM;
        int n = rowBase + r;
        float v = (c < EMB) ? h_in[(size_t)n * EMB + c]
                            : agg[(size_t)n * AGGD + (c - EMB)] * invd[n];
        zt[r * ZPAD + c] = v;
    }
    __syncthreads();

    const int wave = threadIdx.x >> 5;
    const int lane = threadIdx.x & 31;
    const int half = lane >> 4;          // 0: K pair {0,1}, 1: K pair {2,3}
    const int l16  = lane & 15;
    const int koff = half * 2;
    const int col  = wave * 16 + l16;

    const float* zrow0 = &zt[l16 * ZPAD];          // M rows 0..15
    const float* zrow1 = &zt[(l16 + 16) * ZPAD];   // M rows 16..31

    v8f acc0 = {};
    v8f acc1 = {};
    #pragma unroll 4
    for (int k = 0; k < ZDIM; k += 4) {
        v2f a0, a1, b;
        a0.x = zrow0[k + koff];
        a0.y = zrow0[k + koff + 1];
        a1.x = zrow1[k + koff];
        a1.y = zrow1[k + koff + 1];
        b.x = W[(k + koff) * EMB + col];           // one B fragment feeds two WMMAs
        b.y = W[(k + koff + 1) * EMB + col];
        acc0 = __builtin_amdgcn_wmma_f32_16x16x4_f32(
            false, a0, false, b, (short)0, acc0, false, false);
        acc1 = __builtin_amdgcn_wmma_f32_16x16x4_f32(
            false, a1, false, b, (short)0, acc1, false, false);
    }

    // C/D layout: VGPR i -> (M = i + 8*half, N = l16)
    const float bc = bias[col];
    #pragma unroll
    for (int i = 0; i < 8; ++i) {
        int m = i + 8 * half;
        float v0 = acc0[i] + bc;
        float v1 = acc1[i] + bc;
        out[(size_t)(rowBase + m) * EMB + col]      = v0 > 0.0f ? v0 : 0.0f;
        out[(size_t)(rowBase + 16 + m) * EMB + col] = v1 > 0.0f ? v1 : 0.0f;
    }
}

// ---------------- launch ----------------

extern "C" void kernel_launch(void* const* d_in, const int* in_sizes, int n_in,
                              void* d_out, int out_size, void* d_ws, size_t ws_size,
                              hipStream_t stream) {
    const int*   edge_index = (const int*)d_in[0];   // [2, E]
    const float* topic      = (const float*)d_in[1]; // [N, 2]
    const float* h_e        = (const float*)d_in[2]; // [N, 128]
    const float* h_r        = (const float*)d_in[3]; // [E, 128]
    const float* W0         = (const float*)d_in[4]; // [384, 128]
    const float* b0         = (const float*)d_in[5]; // [128]
    const float* W1         = (const float*)d_in[6]; // [384, 128]
    const float* b1         = (const float*)d_in[7]; // [128]

    const int* src = edge_index;
    const int* dst = edge_index + N_EDGES;

    float* out   = (float*)d_out;
    float* out_h = out + (size_t)N_NODES * 2;        // after topic_entity_one_hot

    // workspace: invd[N] (starts as counts) | agg[N*256] | h1[N*128]
    float* invd = (float*)d_ws;
    float* agg  = invd + ((N_NODES + 255) & ~255);
    float* h1   = agg + (size_t)N_NODES * AGGD;

    const int T = 256;
    dim3 blk(T);

    // degree -> inverse denominator (shared by both layers)
    zero_f<<<dim3((N_NODES + T - 1) / T), blk, 0, stream>>>(invd, N_NODES);
    count_deg<<<dim3((N_EDGES + T - 1) / T), blk, 0, stream>>>(invd, dst, N_EDGES);
    inv_denom<<<dim3((N_NODES + T - 1) / T), blk, 0, stream>>>(invd, N_NODES);

    // ---- layer 1 ----
    zero_f<<<dim3((N_NODES * AGGD + T - 1) / T), blk, 0, stream>>>(agg, N_NODES * AGGD);
    scatter_edges<<<dim3(N_EDGES / 4), blk, 0, stream>>>(agg, h_e, h_r, src, dst);
    gemm_bias_relu<<<dim3(N_NODES / MTILE), blk, 0, stream>>>(h_e, agg, invd, W0, b0, h1);

    // ---- layer 2 ----
    zero_f<<<dim3((N_NODES * AGGD + T - 1) / T), blk, 0, stream>>>(agg, N_NODES * AGGD);
    scatter_edges<<<dim3(N_EDGES / 4), blk, 0, stream>>>(agg, h1, h_r, src, dst);
    gemm_bias_relu<<<dim3(N_NODES / MTILE), blk, 0, stream>>>(h1, agg, invd, W1, b1, out_h);

    // first tuple element: topic_entity_one_hot passthrough
    copy_f<<<dim3((N_NODES * 2 + T - 1) / T), blk, 0, stream>>>(out, topic, N_NODES * 2);
}